// SurfacePressurePhysicallyInformedLoss_33346126086163
// MI455X (gfx1250) — compile-verified
//
#include <hip/hip_runtime.h>
#include <hip/hip_bf16.h>
#include <stdint.h>

#define T_DIM 4
#define C_DIM 13
#define H_DIM 361
#define W_DIM 720
#define TILE_W 360
#define HW (H_DIM * W_DIM)
#define THW (T_DIM * HW)

typedef unsigned int u32x4 __attribute__((ext_vector_type(4)));
typedef int          i32x8 __attribute__((ext_vector_type(8)));
typedef int          i32x4 __attribute__((ext_vector_type(4)));

// Issue a 2-D TDM tile load: tile_w x tile_h elements of f32, row stride
// `stride_elems` (elements), from global `gaddr` into LDS byte offset `lds_off`.
__device__ __forceinline__ void tdm_load_2d_f32(uint64_t gaddr, unsigned lds_off,
                                                unsigned tile_w, unsigned tile_h,
                                                unsigned stride_elems) {
    u32x4 g0;
    g0[0] = 1u;                                            // count=1, user-mode D#
    g0[1] = lds_off;                                       // LDS byte address
    g0[2] = (unsigned)(gaddr & 0xFFFFFFFFull);             // global_addr[31:0]
    g0[3] = (unsigned)((gaddr >> 32) & 0x01FFFFFFull)      // global_addr[56:32]
          | (2u << 30);                                    // type = 2 ("image")
    i32x8 g1;
    g1[0] = (int)(2u << 16);                               // data_size=4B, wg_mask=0
    g1[1] = (int)((tile_w & 0xFFFFu) << 16);               // tensor_dim0[15:0]
    g1[2] = (int)(((tile_w >> 16) & 0xFFFFu)               // tensor_dim0[31:16]
          | ((tile_h & 0xFFFFu) << 16));                   // tensor_dim1[15:0]
    g1[3] = (int)(((tile_h >> 16) & 0xFFFFu)               // tensor_dim1[31:16]
          | ((tile_w & 0xFFFFu) << 16));                   // tile_dim0
    g1[4] = (int)(tile_h & 0xFFFFu);                       // tile_dim1 (tile_dim2=0)
    g1[5] = (int)stride_elems;                             // tensor_dim0_stride[31:0]
    g1[6] = 0;                                             // stride0[47:32], dim1_stride lo
    g1[7] = 0;                                             // dim1_stride hi
    i32x4 z4 = {0, 0, 0, 0};                               // groups 2/3 unused (2-D)
    i32x8 z8 = {0, 0, 0, 0, 0, 0, 0, 0};
    __builtin_amdgcn_tensor_load_to_lds(g0, g1, z4, z4, z8, 0);
}

// Pass 1: per (dataset z, t, h, half-row) block. TDM-stage u,v tiles (13 levels x
// 360 w) into LDS, contract over levels with constant dp_c weights, emit
// U = sum_c dp_c*u, V = sum_c dp_c*v, and lsp = log(sp) into workspace.
__global__ __launch_bounds__(256)
void spl_levelreduce(const float* __restrict__ preds_u, const float* __restrict__ preds_v,
                     const float* __restrict__ preds_sp,
                     const float* __restrict__ target_u, const float* __restrict__ target_v,
                     const float* __restrict__ target_sp,
                     const float* __restrict__ atmos_levels,
                     float* __restrict__ Uarr, float* __restrict__ Varr,
                     float* __restrict__ Lsp) {
    __shared__ float smem[2 * C_DIM * TILE_W];   // u tile then v tile
    __shared__ float dps[C_DIM];

    const int wseg = blockIdx.x & 1;            // which half of the row
    const int h    = blockIdx.x >> 1;
    const int t    = blockIdx.y;
    const int z    = blockIdx.z;
    const int tid  = threadIdx.x;

    const float* uptr  = z ? target_u  : preds_u;
    const float* vptr  = z ? target_v  : preds_v;
    const float* spptr = z ? target_sp : preds_sp;

    if (tid < C_DIM) {
        // dp_c: central diff of p_levels = atmos_levels*100 (one-sided at ends)
        float lo = atmos_levels[(tid == 0) ? 0 : tid - 1] * 100.0f;
        float hi = atmos_levels[(tid == C_DIM - 1) ? C_DIM - 1 : tid + 1] * 100.0f;
        float sc = (tid == 0 || tid == C_DIM - 1) ? 1.0f : 0.5f;
        dps[tid] = (hi - lo) * sc;
    }

    if (tid < 32) {  // wave 0 issues the DMA (EXEC ignored; one issue per wave)
        const size_t base = ((size_t)t * C_DIM * H_DIM + (size_t)h) * W_DIM
                          + (size_t)wseg * TILE_W;           // (t, c=0, h, wseg*360)
        tdm_load_2d_f32((uint64_t)(uintptr_t)(uptr + base), 0u,
                        TILE_W, C_DIM, (unsigned)HW);
        tdm_load_2d_f32((uint64_t)(uintptr_t)(vptr + base),
                        (unsigned)(C_DIM * TILE_W * 4),
                        TILE_W, C_DIM, (unsigned)HW);
        __builtin_amdgcn_s_wait_tensorcnt(0);
    }
    __syncthreads();

    const size_t out_base = (((size_t)z * T_DIM + t) * H_DIM + h) * W_DIM
                          + (size_t)wseg * TILE_W;
    const size_t sp_base  = ((size_t)t * H_DIM + h) * W_DIM + (size_t)wseg * TILE_W;

    for (int w = tid; w < TILE_W; w += 256) {
        float U = 0.0f, V = 0.0f;
#pragma unroll
        for (int c = 0; c < C_DIM; ++c) {
            U = fmaf(dps[c], smem[c * TILE_W + w], U);
            V = fmaf(dps[c], smem[C_DIM * TILE_W + c * TILE_W + w], V);
        }
        Uarr[out_base + w] = U;
        Varr[out_base + w] = V;
        Lsp[out_base + w]  = logf(spptr[sp_base + w]);
    }
}

// Pass 2: per-point residual difference + squared error, block partial sums.
__global__ __launch_bounds__(256)
void spl_residual_mse(const float* __restrict__ Uarr, const float* __restrict__ Varr,
                      const float* __restrict__ Lsp,
                      const float* __restrict__ preds_sp, const float* __restrict__ target_sp,
                      const float* __restrict__ lat,
                      const float* __restrict__ dt_p, const float* __restrict__ dx_p,
                      const float* __restrict__ dy_p,
                      float* __restrict__ partials, int npoints) {
    const int p = blockIdx.x * 256 + threadIdx.x;
    float val = 0.0f;
    if (p < npoints) {
        const int t   = p / HW;
        const int rem = p - t * HW;
        const int h   = rem / W_DIM;
        const int w   = rem - h * W_DIM;

        const float m_per_lat = 111194.92664455873f;        // pi*R/180
        const float dt = dt_p[0], dx = dx_p[0], dy = dy_p[0];
        const float ady = dy * m_per_lat;
        const float adx = dx * (m_per_lat * cosf(lat[h] * 0.017453292519943295f));

        float res[2];
#pragma unroll
        for (int z = 0; z < 2; ++z) {
            const float* U  = Uarr + (size_t)z * THW;
            const float* V  = Varr + (size_t)z * THW;
            const float* L  = Lsp  + (size_t)z * THW;
            const float* sp = z ? target_sp : preds_sp;
            const size_t i  = (size_t)t * HW + (size_t)h * W_DIM + (size_t)w;

            float dldt;
            if (t == 0)             dldt = (L[i + HW] - L[i]) / dt;
            else if (t == T_DIM-1)  dldt = (L[i] - L[i - HW]) / dt;
            else                    dldt = (L[i + HW] - L[i - HW]) / (2.0f * dt);

            float dUdx, dLdx;
            if (w == 0)             { dUdx = (U[i+1] - U[i]) / adx;
                                      dLdx = (L[i+1] - L[i]) / adx; }
            else if (w == W_DIM-1)  { dUdx = (U[i] - U[i-1]) / adx;
                                      dLdx = (L[i] - L[i-1]) / adx; }
            else                    { dUdx = (U[i+1] - U[i-1]) / (2.0f * adx);
                                      dLdx = (L[i+1] - L[i-1]) / (2.0f * adx); }

            float dVdy, dLdy;
            if (h == 0)             { dVdy = (V[i+W_DIM] - V[i]) / ady;
                                      dLdy = (L[i+W_DIM] - L[i]) / ady; }
            else if (h == H_DIM-1)  { dVdy = (V[i] - V[i-W_DIM]) / ady;
                                      dLdy = (L[i] - L[i-W_DIM]) / ady; }
            else                    { dVdy = (V[i+W_DIM] - V[i-W_DIM]) / (2.0f * ady);
                                      dLdy = (L[i+W_DIM] - L[i-W_DIM]) / (2.0f * ady); }

            const float invsp = 1.0f / sp[i];
            res[z] = dldt + invsp * (dUdx + dVdy + U[i] * dLdx + V[i] * dLdy);
        }
        const float d = res[0] - res[1];
        val = d * d;
    }

    __shared__ float red[256];
    red[threadIdx.x] = val;
    __syncthreads();
#pragma unroll
    for (int s = 128; s > 0; s >>= 1) {
        if (threadIdx.x < s) red[threadIdx.x] += red[threadIdx.x + s];
        __syncthreads();
    }
    if (threadIdx.x == 0) partials[blockIdx.x] = red[0];
}

// Pass 3: deterministic final reduction, mean, single scalar output.
__global__ __launch_bounds__(256)
void spl_finalize(const float* __restrict__ partials, int n,
                  float* __restrict__ out, float inv_n) {
    __shared__ float red[256];
    float s = 0.0f;
    for (int i = threadIdx.x; i < n; i += 256) s += partials[i];
    red[threadIdx.x] = s;
    __syncthreads();
#pragma unroll
    for (int st = 128; st > 0; st >>= 1) {
        if (threadIdx.x < st) red[threadIdx.x] += red[threadIdx.x + st];
        __syncthreads();
    }
    if (threadIdx.x == 0) out[0] = red[0] * inv_n;
}

extern "C" void kernel_launch(void* const* d_in, const int* in_sizes, int n_in,
                              void* d_out, int out_size, void* d_ws, size_t ws_size,
                              hipStream_t stream) {
    (void)in_sizes; (void)n_in; (void)out_size; (void)ws_size;
    const float* preds_u      = (const float*)d_in[0];
    const float* preds_v      = (const float*)d_in[1];
    const float* preds_sp     = (const float*)d_in[2];
    const float* target_u     = (const float*)d_in[3];
    const float* target_v     = (const float*)d_in[4];
    const float* target_sp    = (const float*)d_in[5];
    const float* lat          = (const float*)d_in[6];
    const float* atmos_levels = (const float*)d_in[7];
    const float* dt           = (const float*)d_in[8];
    const float* dx           = (const float*)d_in[9];
    const float* dy           = (const float*)d_in[10];

    float* ws       = (float*)d_ws;
    float* Uarr     = ws;                 // 2 * THW floats
    float* Varr     = ws + (size_t)2 * THW;
    float* Lsp      = ws + (size_t)4 * THW;
    float* partials = ws + (size_t)6 * THW;

    dim3 g1(H_DIM * 2, T_DIM, 2);
    spl_levelreduce<<<g1, 256, 0, stream>>>(preds_u, preds_v, preds_sp,
                                            target_u, target_v, target_sp,
                                            atmos_levels, Uarr, Varr, Lsp);

    const int npoints = THW;
    const int nblocks = (npoints + 255) / 256;
    spl_residual_mse<<<nblocks, 256, 0, stream>>>(Uarr, Varr, Lsp,
                                                  preds_sp, target_sp, lat,
                                                  dt, dx, dy, partials, npoints);

    spl_finalize<<<1, 256, 0, stream>>>(partials, nblocks, (float*)d_out,
                                        1.0f / (float)npoints);
}